// GPT2Attention_19885698580547
// MI455X (gfx1250) — compile-verified
//
#include <hip/hip_runtime.h>

typedef __attribute__((ext_vector_type(2))) float v2f;
typedef __attribute__((ext_vector_type(8))) float v8f;

__device__ __forceinline__ v8f wmma_f32(v2f a, v2f b, v8f c) {
  // D = A(16x4) * B(4x16) + C(16x16), full fp32 matrix pipe
  return __builtin_amdgcn_wmma_f32_16x16x4_f32(
      /*neg_a=*/false, a, /*neg_b=*/false, b,
      /*c_mod=*/(short)0, c, /*reuse_a=*/false, /*reuse_b=*/false);
}

// ----- CDNA5 async global->LDS copy (ASYNCcnt-tracked, bypasses VGPRs) -----
#if defined(__has_builtin)
#  if __has_builtin(__builtin_amdgcn_global_load_async_to_lds_b128)
#    define ASYNC_B128_BUILTIN 1
#  endif
#  if __has_builtin(__builtin_amdgcn_s_wait_asynccnt)
#    define WAIT_ASYNC_BUILTIN 1
#  endif
#endif

typedef int v4i __attribute__((vector_size(16)));
typedef __attribute__((address_space(1))) v4i gv4i;  // global
typedef __attribute__((address_space(3))) v4i sv4i;  // LDS

__device__ __forceinline__ void async_copy_b128(float* lds_dst, const float* gsrc) {
#ifdef ASYNC_B128_BUILTIN
  float* g = const_cast<float*>(gsrc);
  __builtin_amdgcn_global_load_async_to_lds_b128(
      (gv4i*)g, (sv4i*)lds_dst, 0, 0);
#else
  unsigned lds_off =
      (unsigned)(unsigned long long)(__attribute__((address_space(3))) float*)lds_dst;
  asm volatile("global_load_async_to_lds_b128 %0, %1, off"
               :: "v"(lds_off), "v"(gsrc)
               : "memory");
#endif
}

__device__ __forceinline__ void wait_async0() {
#ifdef WAIT_ASYNC_BUILTIN
  __builtin_amdgcn_s_wait_asynccnt(0);
#else
  asm volatile("s_wait_asynccnt 0x0" ::: "memory");
#endif
}

// ---------------------------------------------------------------------------
// C[M,N] = A[M,K] @ B[K,N] + bias[N]   (M%128==0, N%128==0, K%32==0)
// 256 threads (8 waves). Block tile 128x128, wave tile 64x32.
// Double-buffered async LDS staging.
// ---------------------------------------------------------------------------
#define AST 36   // As row stride (floats): conflict-free for 16-lane b64 reads
#define BST 132  // Bs row stride (floats)

__global__ __launch_bounds__(256) void gemm_bias_f32(
    const float* __restrict__ A, const float* __restrict__ B,
    const float* __restrict__ bias, float* __restrict__ C,
    int M, int N, int K) {
  __shared__ __align__(16) float As[2][128 * AST];  // natural [m][k]
  __shared__ __align__(16) float Bs[2][32 * BST];   // natural [k][n]

  const int tid  = threadIdx.x;
  const int lane = tid & 31;
  const int wave = tid >> 5;
  const int half = lane >> 4;
  const int lm   = lane & 15;
  const int wm   = wave >> 2;   // 0..1
  const int wn   = wave & 3;    // 0..3
  const int rowBase = blockIdx.y * 128;
  const int colBase = blockIdx.x * 128;

  v8f acc[4][2] = {};

  // Issue async prefetch of tile `buf` at depth k0.
  auto prefetch = [&](int buf, int k0) {
#pragma unroll
    for (int i = 0; i < 4; ++i) {  // A: 128 rows x 8 b128 chunks
      int id = tid + 256 * i;
      int r  = id >> 3;
      int c4 = (id & 7) * 4;
      async_copy_b128(&As[buf][r * AST + c4],
                      A + (size_t)(rowBase + r) * K + k0 + c4);
    }
#pragma unroll
    for (int i = 0; i < 4; ++i) {  // B: 32 rows x 32 b128 chunks
      int id = tid + 256 * i;
      int r  = id >> 5;
      int c4 = (id & 31) * 4;
      async_copy_b128(&Bs[buf][r * BST + c4],
                      B + (size_t)(k0 + r) * N + colBase + c4);
    }
  };

  prefetch(0, 0);
  wait_async0();
  __syncthreads();

  const int nk = K >> 5;
  for (int it = 0; it < nk; ++it) {
    const int cur = it & 1;
    if (it + 1 < nk) prefetch(1 - cur, (it + 1) * 32);  // overlap with compute

#pragma unroll
    for (int ks = 0; ks < 32; ks += 4) {
      const int ka = ks + half * 2;  // lanes 0-15: K=ks,ks+1; 16-31: K=ks+2,ks+3
      v2f af[4], bf[2];
#pragma unroll
      for (int am = 0; am < 4; ++am) {
        int m = wm * 64 + am * 16 + lm;
        float2 a2 = *(const float2*)&As[cur][m * AST + ka];  // ds_load_b64
        af[am][0] = a2.x;
        af[am][1] = a2.y;
      }
#pragma unroll
      for (int bn = 0; bn < 2; ++bn) {
        int n = wn * 32 + bn * 16 + lm;
        bf[bn][0] = Bs[cur][ka * BST + n];
        bf[bn][1] = Bs[cur][(ka + 1) * BST + n];
      }
#pragma unroll
      for (int am = 0; am < 4; ++am)
#pragma unroll
        for (int bn = 0; bn < 2; ++bn)
          acc[am][bn] = wmma_f32(af[am], bf[bn], acc[am][bn]);
    }

    wait_async0();     // prefetched tile landed
    __syncthreads();   // all waves done reading `cur` before it is overwritten
  }

#pragma unroll
  for (int am = 0; am < 4; ++am) {
#pragma unroll
    for (int bn = 0; bn < 2; ++bn) {
      int col  = colBase + wn * 32 + bn * 16 + lm;
      float bb = bias[col];
#pragma unroll
      for (int r = 0; r < 8; ++r) {
        int row = rowBase + wm * 64 + am * 16 + r + 8 * half;
        C[(size_t)row * N + col] = acc[am][bn][r] + bb;
      }
    }
  }
}

// ---------------------------------------------------------------------------
// Flash-style causal attention, fp32, H=16, hd=64, S=4096.
// Grid: (S/64, H). Block: 256 threads (8 waves); K/V double-buffered async.
// qkv layout: [S][3072] with q|k|v at col offsets 0/1024/2048 (+ h*64).
// ---------------------------------------------------------------------------
#define QST 68  // row stride (floats) for 64-wide tiles; b64-friendly, conflict-free

__global__ __launch_bounds__(256) void attn_fwd_f32(
    const float* __restrict__ qkv, float* __restrict__ out) {
  const int C3 = 3072, Cout = 1024;
  const int qb = blockIdx.x;
  const int h  = blockIdx.y;

  __shared__ __align__(16) float Qs[64 * QST];     // [q][d]
  __shared__ __align__(16) float Ks[2][64 * QST];  // [key][d]
  __shared__ __align__(16) float Vs[2][64 * QST];  // [key][d]
  __shared__ __align__(16) float Ss[64 * QST];     // [q][key]
  __shared__ float row_m[64];
  __shared__ float row_l[64];
  __shared__ float row_s[64];

  const int tid  = threadIdx.x;
  const int lane = tid & 31;
  const int wave = tid >> 5;
  const int half = lane >> 4;
  const int lm   = lane & 15;
  const int ti   = wave >> 1;        // q sub-tile (0..3)
  const int tj0  = (wave & 1) * 2;   // col sub-tile base (0 or 2)

  auto prefetchKV = [&](int buf, int kb) {
#pragma unroll
    for (int i = 0; i < 4; ++i) {  // 64 rows x 16 b128 chunks each
      int id = tid + 256 * i;
      int r  = id >> 4;
      int c4 = (id & 15) * 4;
      const float* base = qkv + (size_t)(kb * 64 + r) * C3 + h * 64 + c4;
      async_copy_b128(&Ks[buf][r * QST + c4], base + 1024);
      async_copy_b128(&Vs[buf][r * QST + c4], base + 2048);
    }
  };

  // Prefetch Q (unscaled; 1/sqrt(hd) applied to scores) and first K/V block
#pragma unroll
  for (int i = 0; i < 4; ++i) {
    int id = tid + 256 * i;
    int r  = id >> 4;
    int c4 = (id & 15) * 4;
    async_copy_b128(&Qs[r * QST + c4],
                    qkv + (size_t)(qb * 64 + r) * C3 + h * 64 + c4);
  }
  prefetchKV(0, 0);
  if (tid < 64) { row_m[tid] = -1e30f; row_l[tid] = 0.0f; }
  wait_async0();
  __syncthreads();

  v8f oacc[2] = {};

  for (int kb = 0; kb <= qb; ++kb) {
    const int cur = kb & 1;
    if (kb < qb) prefetchKV(1 - cur, kb + 1);  // overlap with this block's math

    // S = Q @ K^T  (each wave: 2 of the 16 16x16 tiles)
    v8f sacc[2] = {};
#pragma unroll
    for (int ks = 0; ks < 64; ks += 4) {
      int ka = ks + half * 2;
      int m  = ti * 16 + lm;
      float2 a2 = *(const float2*)&Qs[m * QST + ka];  // ds_load_b64
      v2f af;
      af[0] = a2.x;
      af[1] = a2.y;
#pragma unroll
      for (int j = 0; j < 2; ++j) {
        int n = (tj0 + j) * 16 + lm;  // key index
        float2 b2 = *(const float2*)&Ks[cur][n * QST + ka];  // B[k][n] = K[n][k]
        v2f bf;
        bf[0] = b2.x;
        bf[1] = b2.y;
        sacc[j] = wmma_f32(af, bf, sacc[j]);
      }
    }
    // Scale + causal mask (diagonal block), scatter to LDS
#pragma unroll
    for (int j = 0; j < 2; ++j) {
      int n = (tj0 + j) * 16 + lm;
#pragma unroll
      for (int r = 0; r < 8; ++r) {
        int m = ti * 16 + r + 8 * half;
        float sv = sacc[j][r] * 0.125f;
        if (kb == qb && n > m) sv += -1.0e9f;
        Ss[m * QST + n] = sv;
      }
    }
    __syncthreads();

    // Online softmax: 4 threads per row, 16 cols each
    {
      int row = tid >> 2;
      int c0  = (tid & 3) * 16;
      float mx = -1e30f;
#pragma unroll
      for (int c = 0; c < 16; ++c) mx = fmaxf(mx, Ss[row * QST + c0 + c]);
      mx = fmaxf(mx, __shfl_xor(mx, 1, 4));
      mx = fmaxf(mx, __shfl_xor(mx, 2, 4));
      float oldm = row_m[row];
      float newm = fmaxf(oldm, mx);
      float lsum = 0.0f;
#pragma unroll
      for (int c = 0; c < 16; ++c) {
        float p = __expf(Ss[row * QST + c0 + c] - newm);
        Ss[row * QST + c0 + c] = p;
        lsum += p;
      }
      lsum += __shfl_xor(lsum, 1, 4);
      lsum += __shfl_xor(lsum, 2, 4);
      if ((tid & 3) == 0) {
        float sc   = __expf(oldm - newm);
        row_s[row] = sc;
        row_m[row] = newm;
        row_l[row] = row_l[row] * sc + lsum;
      }
    }
    __syncthreads();

    // Rescale O, then O += P @ V
#pragma unroll
    for (int j = 0; j < 2; ++j) {
#pragma unroll
      for (int r = 0; r < 8; ++r) {
        int m = ti * 16 + r + 8 * half;
        oacc[j][r] *= row_s[m];
      }
    }
#pragma unroll
    for (int ks = 0; ks < 64; ks += 4) {
      int ka = ks + half * 2;
      int m  = ti * 16 + lm;
      float2 a2 = *(const float2*)&Ss[m * QST + ka];
      v2f af;
      af[0] = a2.x;
      af[1] = a2.y;
#pragma unroll
      for (int j = 0; j < 2; ++j) {
        int n = (tj0 + j) * 16 + lm;  // head-dim index
        v2f bf;
        bf[0] = Vs[cur][ka * QST + n];
        bf[1] = Vs[cur][(ka + 1) * QST + n];
        oacc[j] = wmma_f32(af, bf, oacc[j]);
      }
    }

    wait_async0();     // next K/V block landed
    __syncthreads();   // all waves done with `cur` buffers and Ss
  }

  if (tid < 64) row_s[tid] = 1.0f / row_l[tid];
  __syncthreads();

#pragma unroll
  for (int j = 0; j < 2; ++j) {
    int n = (tj0 + j) * 16 + lm;
#pragma unroll
    for (int r = 0; r < 8; ++r) {
      int m = ti * 16 + r + 8 * half;
      out[(size_t)(qb * 64 + m) * Cout + h * 64 + n] = oacc[j][r] * row_s[m];
    }
  }
}

// ---------------------------------------------------------------------------
extern "C" void kernel_launch(void* const* d_in, const int* in_sizes, int n_in,
                              void* d_out, int out_size, void* d_ws, size_t ws_size,
                              hipStream_t stream) {
  (void)in_sizes; (void)n_in; (void)out_size; (void)ws_size;
  const float* x      = (const float*)d_in[0];
  // d_in[1] = attention_mask (pure causal -1e9; applied analytically in-kernel)
  const float* W_attn = (const float*)d_in[2];
  const float* b_attn = (const float*)d_in[3];
  const float* W_proj = (const float*)d_in[4];
  const float* b_proj = (const float*)d_in[5];
  float* out = (float*)d_out;

  const int S = 4096, C = 1024;
  float* qkv  = (float*)d_ws;             // [4096][3072]  48 MB
  float* attn = qkv + (size_t)S * 3 * C;  // [4096][1024]  16 MB

  dim3 blk(256);
  gemm_bias_f32<<<dim3((3 * C) / 128, S / 128), blk, 0, stream>>>(
      x, W_attn, b_attn, qkv, S, 3 * C, C);
  attn_fwd_f32<<<dim3(S / 64, 16), blk, 0, stream>>>(qkv, attn);
  gemm_bias_f32<<<dim3(C / 128, S / 128), blk, 0, stream>>>(
      attn, W_proj, b_proj, out, S, C, C);
}